// Rgin_77163382440871
// MI455X (gfx1250) — compile-verified
//
#include <hip/hip_runtime.h>
#include <hip/hip_bf16.h>

typedef float v2f __attribute__((ext_vector_type(2)));
typedef float v4f __attribute__((ext_vector_type(4)));
typedef float v8f __attribute__((ext_vector_type(8)));

#define F 128
#define BN_EPS 1e-3f

// ---------------------------------------------------------------------------
// Kernel 1: BatchNorm (inference) + accumulator init
//   h[i,f]   = (x - mean)*rsqrt(var+eps)*gamma + beta        -> d_out (scratch)
//   agg[i,f] = h[i,f] * (coeff_kernel[i] + 1)                -> d_ws
// One wave per node row; each lane owns 4 features. x is read exactly once ->
// non-temporal load keeps it out of L2 (h/agg are the L2 working set).
// ---------------------------------------------------------------------------
__global__ void bn_init_kernel(const float* __restrict__ x,
                               const float* __restrict__ gamma,
                               const float* __restrict__ beta,
                               const float* __restrict__ mean,
                               const float* __restrict__ var,
                               const float* __restrict__ coeff,
                               float* __restrict__ h,
                               float* __restrict__ agg, int N) {
  int tid = blockIdx.x * blockDim.x + threadIdx.x;
  int i = tid >> 5;
  if (i >= N) return;
  int f0 = (tid & 31) << 2;
  float c1 = coeff[i] + 1.0f;
  size_t base = (size_t)i * F + f0;
  v4f xv = __builtin_nontemporal_load((const v4f*)(x + base));
  v4f hv, av;
#pragma unroll
  for (int j = 0; j < 4; ++j) {
    int f = f0 + j;
    float s = gamma[f] * rsqrtf(var[f] + BN_EPS);
    hv[j] = (xv[j] - mean[f]) * s + beta[f];
    av[j] = c1 * hv[j];
  }
  *(v4f*)(h + base) = hv;
  *(v4f*)(agg + base) = av;
}

// ---------------------------------------------------------------------------
// Kernel 2: relation-weighted SpMM scatter:
//   agg[row] += (vals[e]/(rel_coeffs[rel[e]]+1)) * h[col]
// One wave32 per edge. The edge index is wave-uniform: readfirstlane makes it
// an SGPR so the 4 per-edge scalars become s_load broadcasts instead of 32-
// lane vector loads. Each lane owns a float4 slice; fp32 atomics run in L2
// (h + agg = 102 MB, L2-resident in 192 MB).
// ---------------------------------------------------------------------------
__global__ void spmm_scatter_kernel(const int* __restrict__ erow,
                                    const int* __restrict__ ecol,
                                    const float* __restrict__ evals,
                                    const int* __restrict__ erel,
                                    const float* __restrict__ relc,
                                    const float* __restrict__ h,
                                    float* __restrict__ agg, int E) {
  int tid = blockIdx.x * blockDim.x + threadIdx.x;
  int e = __builtin_amdgcn_readfirstlane(tid >> 5);  // wave-uniform edge id
  if (e >= E) return;
  int lane4 = (threadIdx.x & 31) << 2;
  int row = erow[e];
  int col = ecol[e];
  float w = evals[e] / (relc[erel[e]] + 1.0f);
  const float* src = h + (size_t)col * F + lane4;
  float* dst = agg + (size_t)row * F + lane4;
  v4f hv = *(const v4f*)src;
  __hip_atomic_fetch_add(dst + 0, w * hv[0], __ATOMIC_RELAXED, __HIP_MEMORY_SCOPE_AGENT);
  __hip_atomic_fetch_add(dst + 1, w * hv[1], __ATOMIC_RELAXED, __HIP_MEMORY_SCOPE_AGENT);
  __hip_atomic_fetch_add(dst + 2, w * hv[2], __ATOMIC_RELAXED, __HIP_MEMORY_SCOPE_AGENT);
  __hip_atomic_fetch_add(dst + 3, w * hv[3], __ATOMIC_RELAXED, __HIP_MEMORY_SCOPE_AGENT);
}

// ---------------------------------------------------------------------------
// Kernel 3: out = agg @ Wd + bias via V_WMMA_F32_16X16X4_F32 (exact fp32).
// 256 threads = 8 waves/block; each wave owns a 16x128 output strip (8 v8f
// accumulators); block covers 128 rows.
//
// Wd is staged into LDS PRE-SWIZZLED into WMMA B-fragment order:
//   slot(kkIdx, t, lane) holds { B[k0][n], B[k0+1][n] },
//   k0 = 4*kkIdx + 2*(lane>>4), n = 16*t + (lane&15)
// so the inner loop is one conflict-free ds_load_b64 per tile (consecutive
// lanes -> consecutive 8B), with immediate offsets for the 8 tiles.
// A is software-pipelined: next K-step's global_load_b64 is issued before the
// current fragment's 8 WMMAs (index wrapped &127 to stay in-bounds on the
// clamped tail row). EXEC stays all-ones through every WMMA.
// ---------------------------------------------------------------------------
__global__ void gemm_wmma_kernel(const float* __restrict__ A,
                                 const float* __restrict__ Bmat,
                                 const float* __restrict__ bias,
                                 float* __restrict__ out, int N) {
  __shared__ float Bs[F * F];  // 64 KB, fragment-swizzled

  // Stage + swizzle: 32 kkIdx * 8 tiles * 32 lanes = 8192 v2f slots.
  for (int idx = threadIdx.x; idx < 32 * 8 * 32; idx += blockDim.x) {
    int kkIdx = idx >> 8;        // which K-step (4 wide)
    int rest = idx & 255;
    int t = rest >> 5;           // which N-tile
    int l = rest & 31;           // which lane slot
    int k0 = (kkIdx << 2) + ((l >> 4) << 1);
    int n = (t << 4) + (l & 15);
    Bs[idx * 2 + 0] = Bmat[k0 * F + n];
    Bs[idx * 2 + 1] = Bmat[(k0 + 1) * F + n];
  }
  __syncthreads();

  const int wave = threadIdx.x >> 5;
  const int lane = threadIdx.x & 31;
  const int half = lane >> 4;
  const int lm = lane & 15;

  const int mbase = (blockIdx.x * 8 + wave) * 16;
  int mrow = mbase + lm;
  if (mrow >= N) mrow = N - 1;  // clamp loads; keep EXEC all-ones
  const float* __restrict__ Arow = A + (size_t)mrow * F;
  const float* __restrict__ BsLane = Bs + lane * 2;  // 8B-aligned per lane

  v8f acc[8];
#pragma unroll
  for (int t = 0; t < 8; ++t)
    acc[t] = (v8f){0.f, 0.f, 0.f, 0.f, 0.f, 0.f, 0.f, 0.f};

  const int kOff = half << 1;
  v2f a;  // pipelined A fragment
  a[0] = Arow[kOff];
  a[1] = Arow[kOff + 1];

  for (int kkIdx = 0; kkIdx < 32; ++kkIdx) {
    // Issue next A fragment early (wrap keeps the address in-bounds).
    const int kn = ((kkIdx + 1) << 2 | kOff) & 127;
    v2f anext;
    anext[0] = Arow[kn];
    anext[1] = Arow[kn + 1];
    __builtin_prefetch(Arow + ((kn + 16) & 127), 0, 1);  // global_prefetch_b8

    const float* bk = BsLane + (kkIdx << 9);  // 8 tiles * 64 floats per kkIdx
    v2f b[8];
#pragma unroll
    for (int t = 0; t < 8; ++t) b[t] = *(const v2f*)(bk + (t << 6));
#pragma unroll
    for (int t = 0; t < 8; ++t) {
      acc[t] = __builtin_amdgcn_wmma_f32_16x16x4_f32(
          /*neg_a=*/false, a, /*neg_b=*/false, b[t],
          /*c_mod=*/(short)0, acc[t], /*reuse_a=*/false, /*reuse_b=*/false);
    }
    a = anext;
  }

#pragma unroll
  for (int t = 0; t < 8; ++t) {
    const int n = (t << 4) + lm;
    const float bn = bias[n];
#pragma unroll
    for (int v = 0; v < 8; ++v) {
      const int row = mbase + v + (half << 3);
      if (row < N)  // out is written once: non-temporal, bypass L2 pressure
        __builtin_nontemporal_store(acc[t][v] + bn, &out[(size_t)row * F + n]);
    }
  }
}

// ---------------------------------------------------------------------------
// Launch: d_out doubles as scratch for h (overwritten by the final GEMM),
// d_ws holds agg (N*128 floats = 51.2 MB).
// ---------------------------------------------------------------------------
extern "C" void kernel_launch(void* const* d_in, const int* in_sizes, int n_in,
                              void* d_out, int out_size, void* d_ws, size_t ws_size,
                              hipStream_t stream) {
  const float* x     = (const float*)d_in[0];
  const int*   erow  = (const int*)d_in[1];
  const int*   ecol  = (const int*)d_in[2];
  const float* evals = (const float*)d_in[3];
  const int*   erel  = (const int*)d_in[4];
  const float* relc  = (const float*)d_in[5];
  const float* coeff = (const float*)d_in[6];
  const float* Wd    = (const float*)d_in[7];
  const float* bias  = (const float*)d_in[8];
  const float* gamma = (const float*)d_in[9];
  const float* beta  = (const float*)d_in[10];
  const float* mean  = (const float*)d_in[11];
  const float* var   = (const float*)d_in[12];

  const int N = in_sizes[6];   // coeff_kernel has N elements
  const int E = in_sizes[1];   // edge_rows has E elements

  float* h   = (float*)d_out;  // scratch, overwritten by GEMM output
  float* agg = (float*)d_ws;

  // 1) BN + accumulator init: N waves, 32 lanes each.
  {
    long long threads = (long long)N * 32;
    int blocks = (int)((threads + 255) / 256);
    bn_init_kernel<<<blocks, 256, 0, stream>>>(x, gamma, beta, mean, var,
                                               coeff, h, agg, N);
  }

  // 2) Edge scatter-add: E waves, 32 lanes each.
  {
    long long threads = (long long)E * 32;
    int blocks = (int)((threads + 255) / 256);
    spmm_scatter_kernel<<<blocks, 256, 0, stream>>>(erow, ecol, evals, erel,
                                                    relc, h, agg, E);
  }

  // 3) WMMA GEMM: 128 rows per block.
  {
    int blocks = (N + 127) / 128;
    gemm_wmma_kernel<<<blocks, 256, 0, stream>>>(agg, Wd, bias, h /*=out*/, N);
  }
}